// MultiResFeatureGrid2D_88450556494180
// MI455X (gfx1250) — compile-verified
//
#include <hip/hip_runtime.h>
#include <stdint.h>

// ---------------------------------------------------------------------------
// MultiResFeatureGrid2D for MI455X (gfx1250)
//   - levels 0..2 (16^2, 32^2, 64^2 grids; 43 KB total) staged into LDS via
//     the Tensor Data Mover (tensor_load_to_lds + s_wait_tensorcnt)
//   - levels 3..6 gathered straight from L2-resident global grids
//   - nontemporal loads/stores for the streaming coords/output so the 448 MB
//     output stream does not evict the ~11 MB of grid tables from L2
// ---------------------------------------------------------------------------

typedef __attribute__((ext_vector_type(2))) float        f32x2;
typedef __attribute__((ext_vector_type(4))) unsigned int u32x4;
typedef __attribute__((ext_vector_type(4))) int          i32x4;
typedef __attribute__((ext_vector_type(8))) int          i32x8;

#define L0_OFF   0        // floats: 16*16*2  = 512
#define L1_OFF   512      // floats: 32*32*2  = 2048
#define L2_OFF   2560     // floats: 64*64*2  = 8192
#define LDS_F    10752    // total floats (43008 bytes)

#if __has_builtin(__builtin_amdgcn_tensor_load_to_lds)
#define HAVE_TDM 1
#else
#define HAVE_TDM 0
#endif

#if HAVE_TDM
// DMA one full (r x r x 2) f32 grid (contiguous) from global memory into LDS.
// Expressed as a 2D tile: r rows of 2r 4-byte elements, stride 2r.
__device__ __forceinline__ void tdm_grid_to_lds(const float* gsrc,
                                                unsigned lds_byte_addr,
                                                int r) {
    unsigned long long ga = (unsigned long long)(uintptr_t)gsrc;
    unsigned w = 2u * (unsigned)r;  // elements per row (4B each), <= 2048

    u32x4 g0;
    g0.x = 1u;                                     // count=1, user descriptor
    g0.y = lds_byte_addr;                          // LDS byte address
    g0.z = (unsigned)(ga & 0xFFFFFFFFull);         // global_addr[31:0]
    g0.w = (unsigned)((ga >> 32) & 0x01FFFFFFull)  // global_addr[56:32]
         | (2u << 30);                             // type = 2 ("image")

    i32x8 g1;
    g1[0] = (int)(2u << 16);              // data_size=2 -> 4 bytes; mask=0
    g1[1] = (int)(w << 16);               // tensor_dim0[15:0] (bits 63:48)
    g1[2] = (int)(((unsigned)r) << 16);   // dim0 hi16 = 0 | tensor_dim1[15:0]
    g1[3] = (int)(w << 16);               // dim1 hi16 = 0 | tile_dim0 = 2r
    g1[4] = (int)r;                       // tile_dim1 = r, tile_dim2 = 0
    g1[5] = (int)w;                       // tensor_dim0_stride[31:0] = 2r
    g1[6] = 0;                            // stride hi | tensor_dim1_stride lo
    g1[7] = 0;                            // tensor_dim1_stride hi

    i32x4 z4 = {0, 0, 0, 0};
#if __clang_major__ >= 23
    i32x8 z8 = {0, 0, 0, 0, 0, 0, 0, 0};
    __builtin_amdgcn_tensor_load_to_lds(g0, g1, z4, z4, z8, 0);
#else
    __builtin_amdgcn_tensor_load_to_lds(g0, g1, z4, z4, 0);
#endif
}
#endif

// One bilinear lookup of a 2-channel grid at compile-time resolution R.
// x, y are pre-clipped to [0, 1-1e-6], so xs in [0, R-1) and trunc == floor.
template <int R>
__device__ __forceinline__ f32x2 sample_level(const f32x2* __restrict__ g,
                                              float x, float y) {
    const float rm1 = (float)(R - 1);
    float xs = x * rm1;
    float ys = y * rm1;
    int x0 = (int)xs; if (x0 > R - 2) x0 = R - 2;
    int y0 = (int)ys; if (y0 > R - 2) y0 = R - 2;
    float fx = xs - (float)x0;
    float fy = ys - (float)y0;
    int i00 = y0 * R + x0;             // R is a power of two -> shift+add
    f32x2 f00 = g[i00];
    f32x2 f10 = g[i00 + 1];
    f32x2 f01 = g[i00 + R];
    f32x2 f11 = g[i00 + R + 1];
    float gx = 1.0f - fx, gy = 1.0f - fy;
    float w00 = gx * gy, w10 = fx * gy, w01 = gx * fy, w11 = fx * fy;
    f32x2 o;
    o.x = f00.x * w00 + f10.x * w10 + f01.x * w01 + f11.x * w11;
    o.y = f00.y * w00 + f10.y * w10 + f01.y * w01 + f11.y * w11;
    return o;
}

__global__ __launch_bounds__(256)
void mrfg2d_kernel(const f32x2* __restrict__ coords,
                   const float* __restrict__ g0, const float* __restrict__ g1,
                   const float* __restrict__ g2, const float* __restrict__ g3,
                   const float* __restrict__ g4, const float* __restrict__ g5,
                   const float* __restrict__ g6,
                   float* __restrict__ out, int n) {
    __shared__ __align__(16) float s_grid[LDS_F];

#if HAVE_TDM
    if (threadIdx.x == 0) {
        // Low 32 bits of the generic pointer to an LDS object are the
        // workgroup-relative LDS byte address (ISA aperture mapping).
        unsigned base = (unsigned)(uintptr_t)&s_grid[0];
        tdm_grid_to_lds(g0, base + L0_OFF * 4u, 16);
        tdm_grid_to_lds(g1, base + L1_OFF * 4u, 32);
        tdm_grid_to_lds(g2, base + L2_OFF * 4u, 64);
#if __has_builtin(__builtin_amdgcn_s_wait_tensorcnt)
        __builtin_amdgcn_s_wait_tensorcnt(0);
#else
        asm volatile("s_wait_tensorcnt 0x0" ::: "memory");
#endif
    }
    __syncthreads();
#else
    // Fallback: cooperative copy (no TDM builtin on this toolchain).
    for (int i = (int)threadIdx.x; i < 512;  i += (int)blockDim.x) s_grid[L0_OFF + i] = g0[i];
    for (int i = (int)threadIdx.x; i < 2048; i += (int)blockDim.x) s_grid[L1_OFF + i] = g1[i];
    for (int i = (int)threadIdx.x; i < 8192; i += (int)blockDim.x) s_grid[L2_OFF + i] = g2[i];
    __syncthreads();
#endif

    const f32x2* s0 = (const f32x2*)&s_grid[L0_OFF];
    const f32x2* s1 = (const f32x2*)&s_grid[L1_OFF];
    const f32x2* s2 = (const f32x2*)&s_grid[L2_OFF];

    const float hi = 1.0f - 1e-6f;
    int stride = (int)(gridDim.x * blockDim.x);
    for (int pid = (int)(blockIdx.x * blockDim.x + threadIdx.x); pid < n;
         pid += stride) {
        f32x2 c = __builtin_nontemporal_load(&coords[pid]);
        float x = fminf(fmaxf(c.x, 0.0f), hi);
        float y = fminf(fmaxf(c.y, 0.0f), hi);

        f32x2 r0 = sample_level<16>(s0, x, y);     // LDS (ds_load_b64)
        f32x2 r1 = sample_level<32>(s1, x, y);     // LDS
        f32x2 r2 = sample_level<64>(s2, x, y);     // LDS
        f32x2 r3 = sample_level<128>((const f32x2*)g3, x, y);   // L2-resident
        f32x2 r4 = sample_level<256>((const f32x2*)g4, x, y);
        f32x2 r5 = sample_level<512>((const f32x2*)g5, x, y);
        f32x2 r6 = sample_level<1024>((const f32x2*)g6, x, y);

        // 14 floats per point, base offset 56 B -> 8-byte aligned f32x2 stores.
        f32x2* op = (f32x2*)(out + (size_t)pid * 14);
        __builtin_nontemporal_store(r0, op + 0);
        __builtin_nontemporal_store(r1, op + 1);
        __builtin_nontemporal_store(r2, op + 2);
        __builtin_nontemporal_store(r3, op + 3);
        __builtin_nontemporal_store(r4, op + 4);
        __builtin_nontemporal_store(r5, op + 5);
        __builtin_nontemporal_store(r6, op + 6);
    }
}

extern "C" void kernel_launch(void* const* d_in, const int* in_sizes, int n_in,
                              void* d_out, int out_size, void* d_ws, size_t ws_size,
                              hipStream_t stream) {
    (void)n_in; (void)out_size; (void)d_ws; (void)ws_size;
    const f32x2* coords = (const f32x2*)d_in[0];
    const float* g0 = (const float*)d_in[1];
    const float* g1 = (const float*)d_in[2];
    const float* g2 = (const float*)d_in[3];
    const float* g3 = (const float*)d_in[4];
    const float* g4 = (const float*)d_in[5];
    const float* g5 = (const float*)d_in[6];
    const float* g6 = (const float*)d_in[7];
    float* out = (float*)d_out;

    int n = in_sizes[0] / 2;       // B points (coords is B x 2)
    dim3 block(256);
    dim3 grid(2048);               // 16 points/thread; amortizes TDM staging
    mrfg2d_kernel<<<grid, block, 0, stream>>>(coords, g0, g1, g2, g3, g4, g5,
                                              g6, out, n);
}